// Rcell_40192303956631
// MI455X (gfx1250) — compile-verified
//
#include <hip/hip_runtime.h>
#include <stdint.h>

// ---- physical constants folded at compile time (match reference _consts) ----
#define DTs   1e-3f
#define AD    (-0.15f)                 // a  = -0.5*gamma
#define DD    (1.25f)                  // d  = gamma*(nbar+0.5)+lambda
#define CDg   (1.2649110640673518f)    // c  = sqrt(4*eta*lambda)
// Collapsed Riccati constants (C=cI, D=dI => xicov = c*cov + d*I):
//   cov step: p' = E p + F q + G - H (p^2+q^2), etc.,  F = 2*dt*K (runtime)
#define Ecov  (0.9965377223398316f)    // 1 + 2*dt*(a - c*d)
#define Gcov  (-3.125e-4f)             // dt*(d - d^2)
#define Hc2   (1.6e-3f)                // dt*c^2
#define E2m   (0.9982688611699158f)    // 1 + dt*(a - c*d)   (state-map diagonal)

// ---- problem shape (reference: B=8192, T=2048) ----
#define TT        2048
#define TC        64                   // timesteps per chunk (per lane)
#define NC        32                   // chunks per row == lanes per wave
#define WPB       2                    // waves per block in kernel B
#define CHUNK_GB  512                  // bytes of dy per chunk (64*2*4)
#define CHUNK_LB  520                  // padded LDS bytes per chunk (bank-conflict-free)
#define WAVE_LDSF (NC * CHUNK_LB / 4)  // 4160 floats per wave slice

// cov += dt*(A cov + cov A^T + D - xicov xicov^T), collapsed form.
// F = 2*dt*K, FK = dt*K.
__device__ __forceinline__ void cov_step(float& p, float& q, float& r,
                                         float F, float FK) {
  float qq = q * q;
  float s1 = fmaf(p, p, qq);
  float s2 = fmaf(r, r, qq);
  float pr = p + r;
  float np = fmaf(Ecov, p, fmaf(F, q, Gcov)) - Hc2 * s1;
  float nq = fmaf(Ecov, q, FK * (r - p)) - Hc2 * (q * pr);
  float nr = fmaf(Ecov, r, fmaf(-F, q, Gcov)) - Hc2 * s2;
  p = np; q = nq; r = nr;
}

// =====================================================================
// Kernel A: dy-independent covariance pre-scan, checkpoint every TC steps
// =====================================================================
__global__ __launch_bounds__(256)
void cov_ckpt_kernel(const float* __restrict__ cov0,
                     const float* __restrict__ coeffsA,
                     float* __restrict__ ckpt, int B) {
  int b = blockIdx.x * 256 + threadIdx.x;
  if (b >= B) return;
  const float K  = coeffsA[0];
  const float F  = 2.f * DTs * K;
  const float FK = DTs * K;
  const float4 c0 = *(const float4*)(cov0 + (size_t)b * 4);
  float p = c0.x, q = c0.y, r = c0.w;  // c00, c01(=c10), c11
  for (int c = 0; c < NC; ++c) {
    float4 st;
    st.x = p; st.y = q; st.z = r; st.w = 0.f;
    *(float4*)(ckpt + ((size_t)b * NC + c) * 4) = st;
#pragma unroll 8
    for (int i = 0; i < TC; ++i) {
      cov_step(p, q, r, F, FK);
    }
  }
}

// =====================================================================
// Kernel B: one wave32 per batch row.
//   phase 0: async-stage the 16KB dy row into LDS (coalesced b128 bursts, NT)
//   phase 1: per-lane chunk affine-map composition (+ cov replay from ckpt)
//   phase 2: wave-level affine scan via shuffles -> chunk start states
//   phase 3: per-lane replay writing outputs over dy in LDS (in place)
//   phase 4: async-store the 16KB out row from LDS (NT)
// =====================================================================
__global__ __launch_bounds__(32 * WPB)
void filter_scan_kernel(const float* __restrict__ dy,
                        const float* __restrict__ x0,
                        const float* __restrict__ ckpt,
                        const float* __restrict__ coeffsA,
                        float* __restrict__ out, int B) {
  __shared__ float s_lds[WPB * WAVE_LDSF];

  const int wave = threadIdx.x >> 5;
  const int lane = threadIdx.x & 31;
  const int b    = blockIdx.x * WPB + wave;
  if (b >= B) return;

  const float K  = coeffsA[0];
  const float F  = 2.f * DTs * K;
  const float FK = DTs * K;

  // LDS byte address of this wave's slice (flat addr low 32 bits == LDS offset)
  const unsigned sliceByte =
      (unsigned)(uintptr_t)(&s_lds[wave * WAVE_LDSF]);
  const int waveFloatBase = wave * WAVE_LDSF;

  // ---------------- phase 0: async dy row -> LDS ----------------
  {
    unsigned long long rowDy =
        (unsigned long long)(uintptr_t)(dy + (size_t)b * (TT * 2));
#pragma unroll 4
    for (int c = 0; c < NC; ++c) {
      unsigned ldsA = sliceByte + (unsigned)(c * CHUNK_LB + lane * 16);
      unsigned voff = (unsigned)(c * CHUNK_GB + lane * 16);
      asm volatile("global_load_async_to_lds_b128 %0, %1, %2 th:TH_LOAD_NT"
                   :: "v"(ldsA), "v"(voff), "s"(rowDy)
                   : "memory");
    }
  }

  // overlap with async staging: checkpoint cov + row initial state
  const float4 ck = *(const float4*)(ckpt + ((size_t)b * NC + lane) * 4);
  const float xs0 = x0[(size_t)b * 2 + 0];
  const float xs1 = x0[(size_t)b * 2 + 1];

  asm volatile("s_wait_asynccnt 0x0" ::: "memory");

  // ---------------- phase 1: compose chunk affine map ----------------
  const int base = waveFloatBase + lane * (CHUNK_LB / 4);  // 130 floats/chunk
  float p = ck.x, q = ck.y, r = ck.z;
  float M00 = 1.f, M01 = 0.f, M10 = 0.f, M11 = 1.f, V0 = 0.f, V1 = 0.f;
#pragma unroll 4
  for (int i = 0; i < TC; ++i) {
    float dy0 = s_lds[base + 2 * i];
    float dy1 = s_lds[base + 2 * i + 1];
    // step map: x' = S x + v,  S = I + dt*(A - c*xicov) collapsed:
    float s00 = fmaf(-Hc2, p, E2m);
    float s01 = fmaf(-Hc2, q, FK);
    float s10 = fmaf(-Hc2, q, -FK);
    float s11 = fmaf(-Hc2, r, E2m);
    float u = fmaf(CDg, p, DD);
    float w = CDg * q;
    float z = fmaf(CDg, r, DD);
    float v0 = fmaf(u, dy0, w * dy1);
    float v1 = fmaf(w, dy0, z * dy1);
    // acc <- S o acc
    float n00 = fmaf(s00, M00, s01 * M10);
    float n01 = fmaf(s00, M01, s01 * M11);
    float n10 = fmaf(s10, M00, s11 * M10);
    float n11 = fmaf(s10, M01, s11 * M11);
    float nv0 = fmaf(s00, V0, fmaf(s01, V1, v0));
    float nv1 = fmaf(s10, V0, fmaf(s11, V1, v1));
    M00 = n00; M01 = n01; M10 = n10; M11 = n11; V0 = nv0; V1 = nv1;
    cov_step(p, q, r, F, FK);
  }

  // ---------------- phase 2: wave-level affine scan ----------------
#pragma unroll
  for (int d = 1; d < 32; d <<= 1) {
    float o00 = __shfl_up(M00, d, 32);
    float o01 = __shfl_up(M01, d, 32);
    float o10 = __shfl_up(M10, d, 32);
    float o11 = __shfl_up(M11, d, 32);
    float ov0 = __shfl_up(V0, d, 32);
    float ov1 = __shfl_up(V1, d, 32);
    if (lane >= d) {  // acc <- acc o other (other applied first in time)
      float n00 = fmaf(M00, o00, M01 * o10);
      float n01 = fmaf(M00, o01, M01 * o11);
      float n10 = fmaf(M10, o00, M11 * o10);
      float n11 = fmaf(M10, o01, M11 * o11);
      float nv0 = fmaf(M00, ov0, fmaf(M01, ov1, V0));
      float nv1 = fmaf(M10, ov0, fmaf(M11, ov1, V1));
      M00 = n00; M01 = n01; M10 = n10; M11 = n11; V0 = nv0; V1 = nv1;
    }
  }
  // exclusive shift -> transform from t=0 to this chunk's start
  float e00 = __shfl_up(M00, 1, 32);
  float e01 = __shfl_up(M01, 1, 32);
  float e10 = __shfl_up(M10, 1, 32);
  float e11 = __shfl_up(M11, 1, 32);
  float ev0 = __shfl_up(V0, 1, 32);
  float ev1 = __shfl_up(V1, 1, 32);
  if (lane == 0) { e00 = 1.f; e01 = 0.f; e10 = 0.f; e11 = 1.f; ev0 = 0.f; ev1 = 0.f; }

  float X0 = fmaf(e00, xs0, fmaf(e01, xs1, ev0));
  float X1 = fmaf(e10, xs0, fmaf(e11, xs1, ev1));

  // ---------------- phase 3: replay chunk, outputs in place ----------------
  p = ck.x; q = ck.y; r = ck.z;
#pragma unroll 4
  for (int i = 0; i < TC; ++i) {
    float dy0 = s_lds[base + 2 * i];
    float dy1 = s_lds[base + 2 * i + 1];
    // out_t = C x_t dt (pre-update state), overwrite dy slot
    s_lds[base + 2 * i]     = (CDg * DTs) * X0;
    s_lds[base + 2 * i + 1] = (CDg * DTs) * X1;
    float s00 = fmaf(-Hc2, p, E2m);
    float s01 = fmaf(-Hc2, q, FK);
    float s10 = fmaf(-Hc2, q, -FK);
    float s11 = fmaf(-Hc2, r, E2m);
    float u = fmaf(CDg, p, DD);
    float w = CDg * q;
    float z = fmaf(CDg, r, DD);
    float nx0 = fmaf(s00, X0, fmaf(s01, X1, fmaf(u, dy0, w * dy1)));
    float nx1 = fmaf(s10, X0, fmaf(s11, X1, fmaf(w, dy0, z * dy1)));
    X0 = nx0; X1 = nx1;
    cov_step(p, q, r, F, FK);
  }

  // ---------------- phase 4: async out row <- LDS ----------------
  asm volatile("s_wait_dscnt 0x0" ::: "memory");  // LDS writes visible to async engine
  {
    unsigned long long rowOut =
        (unsigned long long)(uintptr_t)(out + (size_t)b * (TT * 2));
#pragma unroll 4
    for (int c = 0; c < NC; ++c) {
      unsigned ldsA = sliceByte + (unsigned)(c * CHUNK_LB + lane * 16);
      unsigned voff = (unsigned)(c * CHUNK_GB + lane * 16);
      asm volatile("global_store_async_from_lds_b128 %0, %1, %2 th:TH_STORE_NT"
                   :: "v"(voff), "v"(ldsA), "s"(rowOut)
                   : "memory");
    }
  }
  asm volatile("s_wait_asynccnt 0x0" ::: "memory");
}

// =====================================================================
extern "C" void kernel_launch(void* const* d_in, const int* in_sizes, int n_in,
                              void* d_out, int out_size, void* d_ws, size_t ws_size,
                              hipStream_t stream) {
  const float* dy     = (const float*)d_in[0];  // [B,T,2]
  const float* x0     = (const float*)d_in[1];  // [B,2]
  const float* cov0   = (const float*)d_in[2];  // [B,2,2]
  const float* coeffs = (const float*)d_in[3];  // [1,1]
  float* out = (float*)d_out;                   // [B,T,2]

  const int B = in_sizes[1] / 2;
  float* ckpt = (float*)d_ws;                   // B*NC*4 floats = 4 MB

  cov_ckpt_kernel<<<dim3((B + 255) / 256), dim3(256), 0, stream>>>(
      cov0, coeffs, ckpt, B);

  filter_scan_kernel<<<dim3((B + WPB - 1) / WPB), dim3(32 * WPB), 0, stream>>>(
      dy, x0, ckpt, coeffs, out, B);
}